// LSTM_13065290515058
// MI455X (gfx1250) — compile-verified
//
#include <hip/hip_runtime.h>
#include <math.h>

// ---------------------------------------------------------------------------
// Problem constants (from reference): T=16384, IN=3, H=100, 4H=400, OUT=1
// ---------------------------------------------------------------------------
#define TSTEPS 16384
#define HDIM   100
#define G4     400   // 4*H

typedef __attribute__((ext_vector_type(2))) float v2f;
typedef __attribute__((ext_vector_type(8))) float v8f;

// ---------------------------------------------------------------------------
// Phase 1: x_gates[t][j] = sum_k X[t][k]*W_ih[j][k] + (b_ih[j]+b_hh[j])
// One wave per 16x16 output tile, single V_WMMA_F32_16X16X4_F32 (K=3 padded
// to 4 with zeros, bias pre-loaded into the C accumulator).
//
// A (16x4 f32, 2 VGPRs): lanes 0-15 -> M=lane, v0=K0, v1=K1
//                        lanes 16-31 -> M=lane-16, v0=K2, v1=K3(=0)
// B mirrors with N across lanes. C/D: 8 VGPRs, N=lane%16, M = r + 8*(lane/16)
// ---------------------------------------------------------------------------
__global__ __launch_bounds__(32) void xgates_wmma_kernel(
    const float* __restrict__ X,     // [T,3]
    const float* __restrict__ Wih,   // [400,3]
    const float* __restrict__ bih,   // [400]
    const float* __restrict__ bhh,   // [400]
    float* __restrict__ xg)          // [T,400]
{
    const int jt   = blockIdx.x;        // 0..24   (column tile, 25*16 = 400)
    const int tt   = blockIdx.y;        // 0..1023 (row tile, 1024*16 = 16384)
    const int lane = threadIdx.x;       // 0..31
    const int half = lane >> 4;         // 0: K=0/1, 1: K=2/3
    const int l    = lane & 15;
    const int row  = tt * 16 + l;       // A row (timestep)
    const int col  = jt * 16 + l;       // B/C column (gate index)

    // Branchless loads so EXEC stays all-ones into the WMMA.
    v2f a, b;
    a.x = X[row * 3 + 2 * half];                 // K0 or K2
    float ay = X[row * 3 + 1];                   // K1 (valid for all lanes)
    a.y = half ? 0.0f : ay;                      // K3 padded to zero
    b.x = Wih[col * 3 + 2 * half];
    float by = Wih[col * 3 + 1];
    b.y = half ? 0.0f : by;

    // Bias depends only on the column -> broadcast into all 8 C rows.
    const float bias = bih[col] + bhh[col];
    v8f c;
#pragma unroll
    for (int r = 0; r < 8; ++r) c[r] = bias;

    // (neg_a, A, neg_b, B, c_mod, C, reuse_a, reuse_b)
    v8f d = __builtin_amdgcn_wmma_f32_16x16x4_f32(
        false, a, false, b, (short)0, c, false, false);

    const int mbase = tt * 16 + half * 8;
#pragma unroll
    for (int r = 0; r < 8; ++r)
        xg[(size_t)(mbase + r) * G4 + col] = d[r];
}

// ---------------------------------------------------------------------------
// Phase 2: sequential LSTM scan. Single persistent workgroup; W_hh (160 KB)
// lives entirely in the 320 KB WGP LDS. 416 threads (13 waves): thread j<400
// owns gate j, threads j<100 own cell state c[j] in a register.
// ---------------------------------------------------------------------------
__global__ __launch_bounds__(416, 1) void lstm_scan_kernel(
    const float* __restrict__ Whh,   // [400,100]
    const float* __restrict__ xg,    // [T,400]
    float* __restrict__ hs)          // [T,100]
{
    __shared__ float sW[G4 * HDIM];  // 160,000 B — fits CDNA5 WGP LDS
    __shared__ float sH[104];        // h (padded for float4 reads)
    __shared__ float sG[G4];         // activated gates

    const int tid = threadIdx.x;

    // Stage W_hh into LDS (coalesced, one time).
    for (int idx = tid; idx < G4 * HDIM; idx += blockDim.x)
        sW[idx] = Whh[idx];
    if (tid < HDIM) sH[tid] = 0.0f;

    float c = 0.0f;                  // cell state, owned by threads < 100
    float xcur = (tid < G4) ? xg[tid] : 0.0f;
    __syncthreads();

    for (int t = 0; t < TSTEPS; ++t) {
        // Software-pipelined load of next step's precomputed gates
        // (independent -> issued before the matvec, waited after).
        float xnext = 0.0f;
        if (tid < G4 && t + 1 < TSTEPS)
            xnext = xg[(size_t)(t + 1) * G4 + tid];

        if (tid < G4) {
            // g[tid] = xg + h . W_hh[tid][:], 25 x (float4 * float4)
            const float4* wr = (const float4*)(sW + tid * HDIM);
            const float4* hv = (const float4*)sH;
            float acc = xcur;
#pragma unroll
            for (int k = 0; k < HDIM / 4; ++k) {
                float4 w4 = wr[k];
                float4 h4 = hv[k];
                acc += w4.x * h4.x + w4.y * h4.y + w4.z * h4.z + w4.w * h4.w;
            }
            // PyTorch gate order: i, f, g, o -> sigmoid/sigmoid/tanh/sigmoid
            float a;
            if (tid >= 200 && tid < 300) a = tanhf(acc);
            else                         a = 1.0f / (1.0f + __expf(-acc));
            sG[tid] = a;
        }
        __syncthreads();

        if (tid < HDIM) {
            float i = sG[tid];
            float f = sG[tid + 100];
            float g = sG[tid + 200];
            float o = sG[tid + 300];
            c = f * c + i * g;
            float h = o * tanhf(c);
            sH[tid] = h;
            hs[(size_t)t * HDIM + tid] = h;   // fire-and-forget store
        }
        __syncthreads();
        xcur = xnext;
    }
}

// ---------------------------------------------------------------------------
// Phase 3: out[t] = sigmoid(hs[t] . W_lin + b_lin). One wave32 per timestep,
// shuffle-based reduction.
// ---------------------------------------------------------------------------
__global__ __launch_bounds__(256) void out_proj_kernel(
    const float* __restrict__ hs,    // [T,100]
    const float* __restrict__ Wlin,  // [100]
    const float* __restrict__ blin,  // [1]
    float* __restrict__ out)         // [T]
{
    const int wave = (blockIdx.x * blockDim.x + threadIdx.x) >> 5;  // = t
    const int lane = threadIdx.x & 31;

    const float* hrow = hs + (size_t)wave * HDIM;
    float s = 0.0f;
    for (int k = lane; k < HDIM; k += 32)
        s += hrow[k] * Wlin[k];
#pragma unroll
    for (int off = 16; off > 0; off >>= 1)
        s += __shfl_down(s, off, 32);
    if (lane == 0)
        out[wave] = 1.0f / (1.0f + __expf(-(s + blin[0])));
}

// ---------------------------------------------------------------------------
// Launch: three kernels on `stream` (serialized by stream order).
// d_ws layout: [0, 26.2MB) x_gates | [26.2MB, 32.8MB) hs
// ---------------------------------------------------------------------------
extern "C" void kernel_launch(void* const* d_in, const int* in_sizes, int n_in,
                              void* d_out, int out_size, void* d_ws, size_t ws_size,
                              hipStream_t stream) {
    const float* X    = (const float*)d_in[0];  // input_x [16384,3]
    const float* Wih  = (const float*)d_in[1];  // W_ih    [400,3]
    const float* Whh  = (const float*)d_in[2];  // W_hh    [400,100]
    const float* bih  = (const float*)d_in[3];  // b_ih    [400]
    const float* bhh  = (const float*)d_in[4];  // b_hh    [400]
    const float* Wlin = (const float*)d_in[5];  // W_lin   [100]
    const float* blin = (const float*)d_in[6];  // b_lin   [1]
    float* out = (float*)d_out;

    float* xg = (float*)d_ws;                        // T*400 floats
    float* hs = xg + (size_t)TSTEPS * G4;            // T*100 floats

    dim3 g1(G4 / 16, TSTEPS / 16);                   // 25 x 1024 tiles
    xgates_wmma_kernel<<<g1, 32, 0, stream>>>(X, Wih, bih, bhh, xg);

    lstm_scan_kernel<<<1, 416, 0, stream>>>(Whh, xg, hs);

    // 8 waves per 256-thread block, one wave per timestep: 2048 blocks exactly.
    out_proj_kernel<<<TSTEPS / 8, 256, 0, stream>>>(hs, Wlin, blin, out);
}